// MPUnoLayer_463856468209
// MI455X (gfx1250) — compile-verified
//
#include <hip/hip_runtime.h>

typedef __attribute__((ext_vector_type(2))) float v2f;
typedef __attribute__((ext_vector_type(8))) float v8f;

#define IN_FEAT 32
#define OUT_FEAT 32

// ---------------------------------------------------------------------------
// Kernel 1: pre-fill output with bias:  out[n][o] = b[o]
// (edge contributions are atomically accumulated on top of this)
// ---------------------------------------------------------------------------
__global__ void bias_init_kernel(const float* __restrict__ bias,
                                 float* __restrict__ out, int total) {
  int i = blockIdx.x * blockDim.x + threadIdx.x;
  if (i < total) out[i] = bias[i & (OUT_FEAT - 1)];
}

// ---------------------------------------------------------------------------
// Kernel 2: Y = feat @ W^T  via V_WMMA_F32_16X16X4_F32 (full fp32 precision).
// One wave computes one 16x16 output tile; K=32 -> 8 chained WMMAs.
//
// A (16x4 f32) layout: lanes 0-15 M=0..15 hold {K=k0,k0+1}; lanes 16-31 hold
// {K=k0+2,k0+3}. B (4x16) mirrored: lane -> column N, half selects K pair.
// D (16x16 f32): c[r] -> row m0 + r + 8*(lane>=16), col n0 + (lane&15).
// ---------------------------------------------------------------------------
__global__ void proj_wmma_kernel(const float* __restrict__ feat,
                                 const float* __restrict__ W,
                                 float* __restrict__ Y, int n_nodes) {
  int wave = (int)((blockIdx.x * blockDim.x + threadIdx.x) >> 5);
  int lane = threadIdx.x & 31;
  int m0 = (wave >> 1) << 4;   // 16-row node tile
  int n0 = (wave & 1) << 4;    // 16-col output-feature tile (OUT_FEAT=32 -> 2)
  if (m0 + 16 > n_nodes) return;   // wave-uniform: EXEC stays all-ones

  int half = lane >> 4;        // 0: K pair (k0,k0+1), 1: K pair (k0+2,k0+3)
  int l15  = lane & 15;
  const float* arow = feat + (size_t)(m0 + l15) * IN_FEAT;
  const float* brow = W    + (size_t)(n0 + l15) * IN_FEAT;  // B[k][n] = W[n][k]

  v8f c = {};
#pragma unroll
  for (int k0 = 0; k0 < IN_FEAT; k0 += 4) {
    int kb = k0 + half * 2;
    v2f a = *(const v2f*)(arow + kb);
    v2f b = *(const v2f*)(brow + kb);
    // 8 args: (neg_a, A, neg_b, B, c_mod, C, reuse_a, reuse_b)
    c = __builtin_amdgcn_wmma_f32_16x16x4_f32(false, a, false, b, (short)0, c,
                                              false, false);
  }
#pragma unroll
  for (int r = 0; r < 8; ++r) {
    Y[(size_t)(m0 + r + half * 8) * OUT_FEAT + n0 + l15] = c[r];
  }
}

// ---------------------------------------------------------------------------
// Kernel 3: edge scatter-add:  out[dst[e]][f] += Y[src[e]][f]
// One wave handles 4 edges (lane = feature). Each edge's gather is a single
// contiguous 128B wave request; 4 edges unrolled for MLP, then 4 native
// fp32 atomics (global_atomic_add_f32, non-returning) into L2-resident out.
// ---------------------------------------------------------------------------
__global__ void scatter_kernel(const float* __restrict__ Y,
                               const int* __restrict__ src,
                               const int* __restrict__ dst,
                               float* __restrict__ out, int n_edges) {
  const int EPW = 4;
  int wave = (int)((blockIdx.x * blockDim.x + threadIdx.x) >> 5);
  int f = threadIdx.x & 31;
  int e0 = wave * EPW;

  float v[EPW];
  int d[EPW];
#pragma unroll
  for (int i = 0; i < EPW; ++i) {
    int e = e0 + i;
    if (e < n_edges) {
      int s = src[e];
      d[i] = dst[e];
      v[i] = Y[(size_t)s * OUT_FEAT + f];
    } else {
      d[i] = -1;
    }
  }
#pragma unroll
  for (int i = 0; i < EPW; ++i) {
    if (d[i] >= 0) {
      unsafeAtomicAdd(out + (size_t)d[i] * OUT_FEAT + f, v[i]);
    }
  }
}

// ---------------------------------------------------------------------------
// Launch: bias fill -> WMMA projection (Y in d_ws) -> atomic edge scatter.
// Inputs (setup_inputs order): feat[f32 N*32], src[i32 E], dst[i32 E],
// W[f32 32*32], b[f32 32]. Output: f32 N*32.
// ---------------------------------------------------------------------------
extern "C" void kernel_launch(void* const* d_in, const int* in_sizes, int n_in,
                              void* d_out, int out_size, void* d_ws, size_t ws_size,
                              hipStream_t stream) {
  const float* feat = (const float*)d_in[0];
  const int*   src  = (const int*)d_in[1];
  const int*   dst  = (const int*)d_in[2];
  const float* W    = (const float*)d_in[3];
  const float* bias = (const float*)d_in[4];

  int n_nodes = in_sizes[0] / IN_FEAT;
  int n_edges = in_sizes[1];
  float* Y   = (float*)d_ws;          // N*OUT_FEAT*4 = 12.8 MB scratch
  float* out = (float*)d_out;

  // 1) out = bias (broadcast over nodes)
  int total = n_nodes * OUT_FEAT;
  bias_init_kernel<<<(total + 255) / 256, 256, 0, stream>>>(bias, out, total);

  // 2) Y = feat @ W^T  (one wave per 16x16 tile; 2 column tiles)
  int tiles  = ((n_nodes + 15) / 16) * 2;           // waves needed
  int blocks = (tiles * 32 + 255) / 256;
  proj_wmma_kernel<<<blocks, 256, 0, stream>>>(feat, W, Y, n_nodes);

  // 3) out[dst] += Y[src]   (one wave per 4 edges)
  int waves   = (n_edges + 3) / 4;
  int sblocks = (waves * 32 + 255) / 256;
  scatter_kernel<<<sblocks, 256, 0, stream>>>(Y, src, dst, out, n_edges);
}